// GNN_10986526343837
// MI455X (gfx1250) — compile-verified
//
#include <hip/hip_runtime.h>
#include <hip/hip_bf16.h>

typedef __attribute__((ext_vector_type(2))) float v2f;
typedef __attribute__((ext_vector_type(8))) float v8f;

#define FD 128   // feature dimension D

// ---------------- zero fill (graph-capture-safe memset) ----------------
__global__ __launch_bounds__(256) void gnn_zero_kernel(float4* __restrict__ p, long n4) {
    long i = (long)blockIdx.x * 256 + threadIdx.x;
    if (i < n4) { float4 z = {0.f, 0.f, 0.f, 0.f}; p[i] = z; }
}

// ---------------- edge scatter: agg[dst] += ew * h[src] ----------------
// one wave32 per edge; each lane owns 4 consecutive features (float4 gather,
// 4x global_atomic_add_f32 scatter). 8 edges per 256-thread block.
__global__ __launch_bounds__(256) void gnn_scatter_kernel(
    const float* __restrict__ h,
    const int*   __restrict__ src,
    const int*   __restrict__ dst,
    const float* __restrict__ ew,
    float*       __restrict__ agg,
    int E)
{
    const int lane = threadIdx.x & 31;
    const int e    = blockIdx.x * 8 + (threadIdx.x >> 5);
    if (e >= E) return;

    const int   s = src[e];
    const int   d = dst[e];
    const float w = ew[e];

    const float4 v = *(const float4*)(h + (size_t)s * FD + lane * 4);
    float* p = agg + (size_t)d * FD + lane * 4;

    __hip_atomic_fetch_add(p + 0, v.x * w, __ATOMIC_RELAXED, __HIP_MEMORY_SCOPE_AGENT);
    __hip_atomic_fetch_add(p + 1, v.y * w, __ATOMIC_RELAXED, __HIP_MEMORY_SCOPE_AGENT);
    __hip_atomic_fetch_add(p + 2, v.z * w, __ATOMIC_RELAXED, __HIP_MEMORY_SCOPE_AGENT);
    __hip_atomic_fetch_add(p + 3, v.w * w, __ATOMIC_RELAXED, __HIP_MEMORY_SCOPE_AGENT);
}

// ---------------- fused dual GEMM + bias + ReLU via f32 WMMA ----------------
// Out = relu(A @ Wr + br + H @ Ws), all fp32, D=128.
// Block = 256 threads = 8 waves = the 8 column tiles of a 64-row stripe.
// Each wave computes FOUR 16x16 output tiles (4 row tiles x its col tile),
// reusing each B (weight) fragment 4x and keeping 4 independent WMMA
// accumulation chains in flight. Row tiles are addressed off ONE base
// pointer per operand with constant 8 KB immediate offsets (no per-tile
// pointers -> low VGPR pressure -> pipelined loads).
// Tail handling: the last block slides back so rows stay in [0, N);
// overlapped rows are recomputed identically by two blocks (benign,
// byte-identical duplicate stores).
__global__ __launch_bounds__(256) void gnn_gemm_kernel(
    const float* __restrict__ A,    // [N][128]  aggregated messages
    const float* __restrict__ H,    // [N][128]  root features
    const float* __restrict__ Wr,   // [128][128] row-major (k, n)
    const float* __restrict__ Ws,   // [128][128]
    const float* __restrict__ br,   // [128]
    float*       __restrict__ Out,  // [N][128]
    int N)
{
    const int lane = threadIdx.x & 31;
    const int wave = threadIdx.x >> 5;           // column tile 0..7
    int rowBase = blockIdx.x * 64;               // 4 row tiles of 16
    if (rowBase + 64 > N) rowBase = N - 64;      // N >= 64 always here
    const int colBase = wave * 16;

    const int mn   = lane & 15;                  // row (A) / col (B,C,D) in tile
    const int kgrp = (lane >> 4) * 2;            // K sub-group: 0 or 2

    const float* ap = A + (size_t)(rowBase + mn) * FD;   // row tile t: +t*2048 floats
    const float* hp = H + (size_t)(rowBase + mn) * FD;
    const float* wr = Wr + colBase + mn;         // row k: +k*128 floats
    const float* ws = Ws + colBase + mn;

    v8f acc0 = {}, acc1 = {}, acc2 = {}, acc3 = {};

#pragma unroll 4
    for (int k = 0; k < FD; k += 4) {
        const int kk = k + kgrp;
        // B fragments: W[kk+j][colBase+n] (shared by all 4 row tiles)
        v2f bw, bs;
        bw.x = wr[(kk)     * FD];
        bw.y = wr[(kk + 1) * FD];
        bs.x = ws[(kk)     * FD];
        bs.y = ws[(kk + 1) * FD];

        // A/H fragments: aligned b64 loads at constant 2048-float tile strides
        v2f a0 = *(const v2f*)(ap + kk);
        v2f a1 = *(const v2f*)(ap + kk + 2048);
        v2f a2 = *(const v2f*)(ap + kk + 4096);
        v2f a3 = *(const v2f*)(ap + kk + 6144);
        v2f h0 = *(const v2f*)(hp + kk);
        v2f h1 = *(const v2f*)(hp + kk + 2048);
        v2f h2 = *(const v2f*)(hp + kk + 4096);
        v2f h3 = *(const v2f*)(hp + kk + 6144);

        acc0 = __builtin_amdgcn_wmma_f32_16x16x4_f32(false, a0, false, bw, (short)0, acc0, false, false);
        acc1 = __builtin_amdgcn_wmma_f32_16x16x4_f32(false, a1, false, bw, (short)0, acc1, false, false);
        acc2 = __builtin_amdgcn_wmma_f32_16x16x4_f32(false, a2, false, bw, (short)0, acc2, false, false);
        acc3 = __builtin_amdgcn_wmma_f32_16x16x4_f32(false, a3, false, bw, (short)0, acc3, false, false);
        acc0 = __builtin_amdgcn_wmma_f32_16x16x4_f32(false, h0, false, bs, (short)0, acc0, false, false);
        acc1 = __builtin_amdgcn_wmma_f32_16x16x4_f32(false, h1, false, bs, (short)0, acc1, false, false);
        acc2 = __builtin_amdgcn_wmma_f32_16x16x4_f32(false, h2, false, bs, (short)0, acc2, false, false);
        acc3 = __builtin_amdgcn_wmma_f32_16x16x4_f32(false, h3, false, bs, (short)0, acc3, false, false);
    }

    // Epilogue: bias + ReLU. All rows in-bounds by construction; stores use
    // constant offsets off one base pointer.
    const float bias = br[colBase + mn];
    const int hi16 = 8 * (lane >> 4);            // C/D layout: row = i + 8*(lane>=16)
    float* op = Out + (size_t)(rowBase + hi16) * FD + colBase + mn;
    v8f accs[4] = {acc0, acc1, acc2, acc3};
#pragma unroll
    for (int t = 0; t < 4; ++t) {
#pragma unroll
        for (int i = 0; i < 8; ++i) {
            float v = accs[t][i] + bias;
            op[(t * 16 + i) * FD] = v > 0.f ? v : 0.f;
        }
    }
}

// ---------------- driver ----------------
extern "C" void kernel_launch(void* const* d_in, const int* in_sizes, int n_in,
                              void* d_out, int out_size, void* d_ws, size_t ws_size,
                              hipStream_t stream) {
    const float* x   = (const float*)d_in[0];
    const int*   ei  = (const int*)  d_in[1];   // edge_index [2][E]
    const float* ew  = (const float*)d_in[2];
    const float* Wr0 = (const float*)d_in[3];
    const float* br0 = (const float*)d_in[4];
    const float* Ws0 = (const float*)d_in[5];
    const float* Wr1 = (const float*)d_in[6];
    const float* br1 = (const float*)d_in[7];
    const float* Ws1 = (const float*)d_in[8];

    const int N = in_sizes[0] / FD;   // 100000
    const int E = in_sizes[2];        // 1600000
    const int* src = ei;
    const int* dst = ei + E;

    float* agg = (float*)d_ws;                       // N*128 floats (51.2 MB)
    float* h1  = agg + (size_t)N * FD;               // N*128 floats (51.2 MB)
    float* out = (float*)d_out;

    const long n4      = (long)N * FD / 4;
    const int  zeroGrd = (int)((n4 + 255) / 256);
    const int  scatGrd = (E + 7) / 8;
    const int  gemmGrd = (N + 63) / 64;

    // Layer 0
    gnn_zero_kernel   <<<zeroGrd, 256, 0, stream>>>((float4*)agg, n4);
    gnn_scatter_kernel<<<scatGrd, 256, 0, stream>>>(x, src, dst, ew, agg, E);
    gnn_gemm_kernel   <<<gemmGrd, 256, 0, stream>>>(agg, x, Wr0, Ws0, br0, h1, N);
    // Layer 1
    gnn_zero_kernel   <<<zeroGrd, 256, 0, stream>>>((float4*)agg, n4);
    gnn_scatter_kernel<<<scatGrd, 256, 0, stream>>>(h1, src, dst, ew, agg, E);
    gnn_gemm_kernel   <<<gemmGrd, 256, 0, stream>>>(agg, h1, Wr1, Ws1, br1, out, N);
}